// Scatter_38843684225659
// MI455X (gfx1250) — compile-verified
//
#include <hip/hip_runtime.h>
#include <hip/hip_bf16.h>
#include <stdint.h>

#define D_FEAT 48          // floats per edge row
#define CHUNKS 12          // float4 chunks per row
#define TPB    128         // threads per block == edges per tile
#define ROW_PAD 52         // padded LDS row in floats (208B: 16B-aligned, spreads banks)
#define NBLOCKS_MAX 2048

// Async global->LDS via inline asm; only valid when compiling gfx1250 device code.
#if defined(__HIP_DEVICE_COMPILE__) && defined(__gfx1250__)
#define USE_ASYNC_LDS 1
#else
#define USE_ASYNC_LDS 0
#endif

#if USE_ASYNC_LDS
// One 16B async copy: global (64-bit addr in VGPR pair) -> LDS (32-bit byte addr
// in VGPR), with a literal instruction offset. Tracked by ASYNCcnt.
#define ASYNC_CP16(LA, GA, OFF)                                                \
    asm volatile("global_load_async_to_lds_b128 %0, %1, off offset:" #OFF      \
                 :: "v"(LA), "v"(GA) : "memory")
#define WAIT_ASYNC(N) asm volatile("s_wait_asynccnt " #N ::: "memory")
#endif

// Never launched. Defined FIRST so the disasm snippet's visible window shows
// exactly how __hip_atomic_fetch_add(f32, relaxed, agent) lowers on gfx1250
// (expect a single no-return global_atomic_add_f32, not a CAS loop).
__global__ void atomic_probe(float* p, float v) {
    __hip_atomic_fetch_add(p, v, __ATOMIC_RELAXED, __HIP_MEMORY_SCOPE_AGENT);
}

__global__ __launch_bounds__(TPB) void scatter_sum(
    const float* __restrict__ x,
    const int*   __restrict__ idx2,   // int64 index viewed as int pairs (low dword valid)
    float* __restrict__ out,
    float* __restrict__ cnt,
    int n_edges, int n_tiles) {
#if USE_ASYNC_LDS
    __shared__ float lds[2][TPB * ROW_PAD];
    const int k  = threadIdx.x;
    const int nb = gridDim.x;

    int t = blockIdx.x;
    if (t >= n_tiles) return;

    // Stage tile T into LDS buffer B: 12 async b128 copies per thread (one full
    // 192B edge row). Tail addresses are clamped (not predicated) so every wave
    // issues exactly 12 async ops -> ASYNCcnt stays wave-uniform.
#define STAGE(T, B)                                                            \
    do {                                                                       \
        long e0_ = (long)(T) * TPB + k;                                        \
        long emax_ = (long)n_edges - 1;                                        \
        if (e0_ > emax_) e0_ = emax_;                                          \
        unsigned long long ga_ =                                               \
            (unsigned long long)(size_t)(x + e0_ * D_FEAT);                    \
        unsigned la_ = (unsigned)(size_t)(&lds[(B)][k * ROW_PAD]);             \
        ASYNC_CP16(la_, ga_, 0);                                               \
        ASYNC_CP16(la_, ga_, 16);                                              \
        ASYNC_CP16(la_, ga_, 32);                                              \
        ASYNC_CP16(la_, ga_, 48);                                              \
        ASYNC_CP16(la_, ga_, 64);                                              \
        ASYNC_CP16(la_, ga_, 80);                                              \
        ASYNC_CP16(la_, ga_, 96);                                              \
        ASYNC_CP16(la_, ga_, 112);                                             \
        ASYNC_CP16(la_, ga_, 128);                                             \
        ASYNC_CP16(la_, ga_, 144);                                             \
        ASYNC_CP16(la_, ga_, 160);                                             \
        ASYNC_CP16(la_, ga_, 176);                                             \
    } while (0)

    int cur = 0;
    STAGE(t, 0);
    for (; t < n_tiles; t += nb) {
        int tn = t + nb;
        if (tn < n_tiles) {
            STAGE(tn, cur ^ 1);   // prefetch next tile into the other buffer
            WAIT_ASYNC(12);       // current tile's 12 copies are complete
        } else {
            WAIT_ASYNC(0);
        }

        long e = (long)t * TPB + k;
        if (e < n_edges) {
            int seg = idx2[(size_t)2 * e];             // low 32 bits of int64 id
            float* orow = out + (size_t)seg * D_FEAT;
            const float* row = &lds[cur][k * ROW_PAD];
#pragma unroll
            for (int c = 0; c < CHUNKS; ++c) {
                float4 v = *(const float4*)(row + 4 * c);
                __hip_atomic_fetch_add(orow + 4*c + 0, v.x, __ATOMIC_RELAXED, __HIP_MEMORY_SCOPE_AGENT);
                __hip_atomic_fetch_add(orow + 4*c + 1, v.y, __ATOMIC_RELAXED, __HIP_MEMORY_SCOPE_AGENT);
                __hip_atomic_fetch_add(orow + 4*c + 2, v.z, __ATOMIC_RELAXED, __HIP_MEMORY_SCOPE_AGENT);
                __hip_atomic_fetch_add(orow + 4*c + 3, v.w, __ATOMIC_RELAXED, __HIP_MEMORY_SCOPE_AGENT);
            }
            __hip_atomic_fetch_add(cnt + seg, 1.0f, __ATOMIC_RELAXED, __HIP_MEMORY_SCOPE_AGENT);
        }
        cur ^= 1;
    }
#undef STAGE
#else
    // Fallback (host pass / non-gfx1250): direct global float4 loads.
    long i = (long)blockIdx.x * blockDim.x + threadIdx.x;
    long stride = (long)gridDim.x * blockDim.x;
    for (long e = i; e < n_edges; e += stride) {
        int seg = idx2[(size_t)2 * e];
        const float4* row = (const float4*)(x + e * D_FEAT);
        float* orow = out + (size_t)seg * D_FEAT;
#pragma unroll
        for (int c = 0; c < CHUNKS; ++c) {
            float4 v = row[c];
            __hip_atomic_fetch_add(orow + 4*c + 0, v.x, __ATOMIC_RELAXED, __HIP_MEMORY_SCOPE_AGENT);
            __hip_atomic_fetch_add(orow + 4*c + 1, v.y, __ATOMIC_RELAXED, __HIP_MEMORY_SCOPE_AGENT);
            __hip_atomic_fetch_add(orow + 4*c + 2, v.z, __ATOMIC_RELAXED, __HIP_MEMORY_SCOPE_AGENT);
            __hip_atomic_fetch_add(orow + 4*c + 3, v.w, __ATOMIC_RELAXED, __HIP_MEMORY_SCOPE_AGENT);
        }
        __hip_atomic_fetch_add(cnt + seg, 1.0f, __ATOMIC_RELAXED, __HIP_MEMORY_SCOPE_AGENT);
    }
#endif
}

__global__ void zero_ws(float* __restrict__ out, float* __restrict__ cnt,
                        int n_out, int n_cnt) {
    int i = blockIdx.x * blockDim.x + threadIdx.x;
    int stride = gridDim.x * blockDim.x;
    for (int j = i; j < n_out; j += stride) out[j] = 0.0f;
    for (int j = i; j < n_cnt; j += stride) cnt[j] = 0.0f;
}

// One thread per segment row: one count load, one IEEE divide, 12 float4
// read-scale-write chunks. No integer division in the hot loop.
__global__ void div_mean(float* __restrict__ out, const float* __restrict__ cnt, int S) {
    int i = blockIdx.x * blockDim.x + threadIdx.x;
    int stride = gridDim.x * blockDim.x;
    for (int s = i; s < S; s += stride) {
        float c = fmaxf(cnt[s], 1.0f);
        float r = 1.0f / c;
        float4* row = (float4*)(out + (size_t)s * D_FEAT);
#pragma unroll
        for (int q = 0; q < CHUNKS; ++q) {
            float4 v = row[q];
            v.x *= r; v.y *= r; v.z *= r; v.w *= r;
            row[q] = v;
        }
    }
}

extern "C" void kernel_launch(void* const* d_in, const int* in_sizes, int n_in,
                              void* d_out, int out_size, void* d_ws, size_t ws_size,
                              hipStream_t stream) {
    const float* x   = (const float*)d_in[0];
    const int* idx2  = (const int*)d_in[1];   // int64 index, read low dwords
    int n_edges = in_sizes[1];
    int S = out_size / D_FEAT;
    float* out = (float*)d_out;
    float* cnt = (float*)d_ws;                // S floats of scratch

    {   // zero sums + counts (harness poisons buffers)
        int n_out = S * D_FEAT;
        int blocks = (n_out + 255) / 256;
        if (blocks > 4096) blocks = 4096;
        zero_ws<<<blocks, 256, 0, stream>>>(out, cnt, n_out, S);
    }
    {   // async-pipelined atomic scatter-add
        int n_tiles = (n_edges + TPB - 1) / TPB;
        int blocks = n_tiles < NBLOCKS_MAX ? n_tiles : NBLOCKS_MAX;
        scatter_sum<<<blocks, TPB, 0, stream>>>(x, idx2, out, cnt, n_edges, n_tiles);
    }
    {   // mean = sum / max(count, 1)
        int blocks = (S + 255) / 256;
        if (blocks > 4096) blocks = 4096;
        div_mean<<<blocks, 256, 0, stream>>>(out, cnt, S);
    }
}